// RadixAttention_89300960019073
// MI455X (gfx1250) — compile-verified
//
#include <hip/hip_runtime.h>
#include <math.h>

// ---- problem constants (match reference) ----
#define NUM_HEADS    32
#define NUM_KV_HEADS 8
#define GQA_G        4          // NUM_HEADS / NUM_KV_HEADS
#define HEAD_DIM     128
#define WINDOW       1024
#define SCALING      0.088388347648318447f   // 1/sqrt(128)
#define LOGIT_CAP    30.0f
#define INV_CAP      (1.0f / 30.0f)

#define WAVES        8
#define TILE         16
#define NEG_BIG      (-1.0e30f)

typedef __attribute__((ext_vector_type(2))) float v2f;
typedef __attribute__((ext_vector_type(8))) float v8f;

// One workgroup per (batch, kv_head). 8 wave32s; each wave strides over
// 16-key tiles of the attention window with a private online softmax,
// using V_WMMA_F32_16X16X4_F32 for both Q*K^T and P*V. Cross-wave merge
// via LDS at the end (classic flash-attention combine).
__global__ __launch_bounds__(256)
void radix_attn_f32wmma_kernel(const float* __restrict__ q,
                               const float* __restrict__ k_buffer,
                               const float* __restrict__ v_buffer,
                               const int*   __restrict__ req_to_token,
                               const int*   __restrict__ seq_lens,
                               float*       __restrict__ out,
                               int max_len)
{
    const int b    = blockIdx.x;        // batch
    const int kh   = blockIdx.y;        // kv head
    const int tid  = threadIdx.x;
    const int wave = tid >> 5;
    const int lane = tid & 31;
    const int col  = lane & 15;         // N index within tile / M index for A
    const int half = lane >> 4;         // half-wave selector for K pairs

    __shared__ float lds_p[WAVES][16][16];            // P re-fragment bounce
    __shared__ float lds_o[WAVES][GQA_G][HEAD_DIM];   // per-wave partial O
    __shared__ float lds_m[WAVES][GQA_G];             // per-wave row max
    __shared__ float lds_l[WAVES][GQA_G];             // per-wave row sumexp

    const int sl = seq_lens[b];
    int w0 = sl - WINDOW; if (w0 < 0) w0 = 0;
    const int W = sl - w0;                            // valid window length (>=1)

    const int* rtt = req_to_token + (size_t)b * max_len + w0;

    // ---------------- Q A-fragments, resident for the whole key loop ----
    // A (16x4 f32): lane l holds row M=l&15, K = 4*kc + 2*half + {0,1}
    v2f aq[32];
    {
        const int m = col;
        if (m < GQA_G) {
            const float* qrow = q + ((size_t)b * NUM_HEADS + (size_t)kh * GQA_G + m) * HEAD_DIM;
            #pragma unroll
            for (int kc = 0; kc < 32; ++kc) {
                const int d = 4 * kc + 2 * half;
                aq[kc].x = qrow[d];
                aq[kc].y = qrow[d + 1];
            }
        } else {
            #pragma unroll
            for (int kc = 0; kc < 32; ++kc) { aq[kc].x = 0.0f; aq[kc].y = 0.0f; }
        }
    }

    // ---------------- accumulators -------------------------------------
    v8f o[8];                       // O: 16 rows x 128 dims as 8 C-tiles
    #pragma unroll
    for (int t = 0; t < 8; ++t)
        #pragma unroll
        for (int r = 0; r < 8; ++r) o[t][r] = 0.0f;

    float row_m[8], row_l[8];       // per-lane rows r + 8*half
    #pragma unroll
    for (int r = 0; r < 8; ++r) { row_m[r] = NEG_BIG; row_l[r] = 0.0f; }

    const int ntiles = (W + TILE - 1) / TILE;

    for (int tb = wave; tb < ntiles; tb += WAVES) {
        // ---- token for my column ----
        const int  mypos   = tb * TILE + col;
        const bool myvalid = (mypos < W);
        const int  mytok   = rtt[myvalid ? mypos : 0];
        const float* krow  = k_buffer + ((size_t)mytok * NUM_KV_HEADS + kh) * HEAD_DIM;

        // ---- S = Q * K^T over D=128, chained 16x16x4 f32 WMMAs ----
        v8f s;
        #pragma unroll
        for (int r = 0; r < 8; ++r) s[r] = 0.0f;

        #pragma unroll
        for (int kc = 0; kc < 32; ++kc) {
            const int d = 4 * kc + 2 * half;
            v2f bk;
            bk.x = krow[d];
            bk.y = krow[d + 1];
            s = __builtin_amdgcn_wmma_f32_16x16x4_f32(
                    false, aq[kc], false, bk, (short)0, s, false, false);
        }

        // ---- scale + tanh logit cap + window mask ----
        #pragma unroll
        for (int r = 0; r < 8; ++r) {
            float sv = s[r] * SCALING;
            sv = LOGIT_CAP * tanhf(sv * INV_CAP);
            s[r] = myvalid ? sv : NEG_BIG;
        }

        // ---- online softmax (row = r + 8*half lives in one 16-lane half) ----
        float corr[8];
        #pragma unroll
        for (int r = 0; r < 8; ++r) {
            float v = s[r];
            v = fmaxf(v, __shfl_xor(v, 1));
            v = fmaxf(v, __shfl_xor(v, 2));
            v = fmaxf(v, __shfl_xor(v, 4));
            v = fmaxf(v, __shfl_xor(v, 8));      // tile row max

            const float nm = fmaxf(row_m[r], v);
            const float c  = __expf(row_m[r] - nm);
            float pv = myvalid ? __expf(s[r] - nm) : 0.0f;

            lds_p[wave][r + 8 * half][col] = pv;  // stash P[M][N]

            float ss = pv;
            ss += __shfl_xor(ss, 1);
            ss += __shfl_xor(ss, 2);
            ss += __shfl_xor(ss, 4);
            ss += __shfl_xor(ss, 8);              // tile row sum

            row_l[r] = row_l[r] * c + ss;
            row_m[r] = nm;
            corr[r]  = c;
        }

        // rescale accumulator rows by correction factors
        #pragma unroll
        for (int t = 0; t < 8; ++t)
            #pragma unroll
            for (int r = 0; r < 8; ++r) o[t][r] *= corr[r];

        __builtin_amdgcn_wave_barrier();          // order LDS store -> load (same wave)

        // ---- re-fragment P into A layout from LDS ----
        v2f ap[4];
        #pragma unroll
        for (int kc2 = 0; kc2 < 4; ++kc2) {
            const int kk = 4 * kc2 + 2 * half;
            ap[kc2].x = lds_p[wave][col][kk];
            ap[kc2].y = lds_p[wave][col][kk + 1];
        }

        // ---- O += P * V, 16-key K-dim as 4 chained K=4 WMMAs x 8 dim tiles ----
        #pragma unroll
        for (int kc2 = 0; kc2 < 4; ++kc2) {
            const int k0 = 4 * kc2 + 2 * half;    // key rows of the B fragment
            const int p0 = tb * TILE + k0;
            const int p1 = p0 + 1;
            const int t0 = rtt[(p0 < W) ? p0 : 0];
            const int t1 = rtt[(p1 < W) ? p1 : 0];
            const float* v0 = v_buffer + ((size_t)t0 * NUM_KV_HEADS + kh) * HEAD_DIM;
            const float* v1 = v_buffer + ((size_t)t1 * NUM_KV_HEADS + kh) * HEAD_DIM;
            #pragma unroll
            for (int t = 0; t < 8; ++t) {
                v2f bv;
                bv.x = v0[t * 16 + col];
                bv.y = v1[t * 16 + col];
                o[t] = __builtin_amdgcn_wmma_f32_16x16x4_f32(
                           false, ap[kc2], false, bv, (short)0, o[t], false, false);
            }
        }

        __builtin_amdgcn_wave_barrier();          // before next tile reuses lds_p
    }

    // ---------------- export per-wave partials (rows 0..3 only) --------
    if (half == 0) {
        #pragma unroll
        for (int r = 0; r < GQA_G; ++r) {
            #pragma unroll
            for (int t = 0; t < 8; ++t)
                lds_o[wave][r][t * 16 + col] = o[t][r];
            if (col == 0) { lds_m[wave][r] = row_m[r]; lds_l[wave][r] = row_l[r]; }
        }
    }
    __syncthreads();

    // ---------------- cross-wave flash-attention merge ------------------
    float* outp = out + ((size_t)b * NUM_HEADS + (size_t)kh * GQA_G) * HEAD_DIM;
    for (int e = tid; e < GQA_G * HEAD_DIM; e += blockDim.x) {
        const int r = e >> 7;            // row 0..3
        const int d = e & 127;           // dim
        float M = NEG_BIG;
        #pragma unroll
        for (int w = 0; w < WAVES; ++w) M = fmaxf(M, lds_m[w][r]);
        float acc = 0.0f, lsum = 0.0f;
        #pragma unroll
        for (int w = 0; w < WAVES; ++w) {
            const float sc = __expf(lds_m[w][r] - M);
            acc  += sc * lds_o[w][r][d];
            lsum += sc * lds_l[w][r];
        }
        outp[(size_t)r * HEAD_DIM + d] = acc / lsum;
    }
}

extern "C" void kernel_launch(void* const* d_in, const int* in_sizes, int n_in,
                              void* d_out, int out_size, void* d_ws, size_t ws_size,
                              hipStream_t stream) {
    const float* q   = (const float*)d_in[0];
    const float* kb  = (const float*)d_in[1];
    const float* vb  = (const float*)d_in[2];
    const int*   rtt = (const int*)d_in[3];
    const int*   sl  = (const int*)d_in[4];
    float*       out = (float*)d_out;

    const int B       = in_sizes[0] / (NUM_HEADS * HEAD_DIM);
    const int max_len = in_sizes[3] / B;

    dim3 grid(B, NUM_KV_HEADS);
    radix_attn_f32wmma_kernel<<<grid, 256, 0, stream>>>(q, kb, vb, rtt, sl, out, max_len);
}